// CausalConv1d_46033459478752
// MI455X (gfx1250) — compile-verified
//
#include <hip/hip_runtime.h>

// Causal depthwise conv1d, K=4, x:(8,8192,1024) f32 -> y same shape.
// Memory-bound (AI ~ 1 FLOP/B): floor = 537MB / 23.3TB/s ~ 23us.
// CDNA5 path: GLOBAL_LOAD_ASYNC_TO_LDS_B128 ring (ASYNCcnt), ds_load_b128
// consume, register sliding window, nontemporal global_store_b128.
// Steady-state loop is unrolled x8 so every LDS slot / global row offset is
// an instruction immediate (offset: field), minimizing VALU/SALU overhead.

#define FEATURE_DIM 1024
#define SEQ_LEN     8192
#define BATCH       8
#define F4          (FEATURE_DIM / 4)   // 256 float4 groups == blockDim.x
#define T_TILE      64                  // timesteps per block
#define DEPTH       8                   // LDS ring slots (8 * 4KB = 32KB)
#define ROW_BYTES   (FEATURE_DIM * 4)   // 4096 bytes per timestep row

#if __has_builtin(__builtin_amdgcn_s_wait_asynccnt)
#define WAIT_ASYNCCNT(n) __builtin_amdgcn_s_wait_asynccnt(n)
#else
#define WAIT_ASYNCCNT(n) asm volatile("s_wait_asynccnt %0" :: "i"(n) : "memory")
#endif

#if __has_builtin(__builtin_amdgcn_s_wait_dscnt)
#define WAIT_DSCNT(n) __builtin_amdgcn_s_wait_dscnt(n)
#else
#define WAIT_DSCNT(n) asm volatile("s_wait_dscnt %0" :: "i"(n) : "memory")
#endif

typedef float vfloat4 __attribute__((ext_vector_type(4)));

// GLOBAL_LOAD_ASYNC_TO_LDS_B128, GVS mode with immediate row offset:
//   lds[VGPR ldsoff] <- mem[SGPR64 base + VGPR32 goff + imm]
// Per-lane 16B; per wave32 512 contiguous bytes.
__device__ __forceinline__ void async_row_to_lds(unsigned long long gbase,
                                                 unsigned goff, unsigned ldsoff,
                                                 int imm) {
  asm volatile("global_load_async_to_lds_b128 %0, %1, %2 offset:%3"
               :
               : "v"(ldsoff), "v"(goff), "s"(gbase), "i"(imm)
               : "memory");
}

__device__ __forceinline__ float4 fma4(float4 w, float4 a, float4 acc) {
  acc.x += w.x * a.x;
  acc.y += w.y * a.y;
  acc.z += w.z * a.z;
  acc.w += w.w * a.w;
  return acc;
}

__device__ __forceinline__ void store_nt(float4* p, float4 o) {
  vfloat4 v = {o.x, o.y, o.z, o.w};
  __builtin_nontemporal_store(v, (vfloat4*)p);
}

__global__ __launch_bounds__(F4) void causal_dwconv4_kernel(
    const float* __restrict__ x, const float* __restrict__ W,
    const float* __restrict__ bias, float* __restrict__ y) {
  __shared__ float4 lds[DEPTH * F4];  // 8 rows x 4KB

  const int tid = threadIdx.x;              // feature float4 group, 0..255
  const int t0  = blockIdx.x * T_TILE;      // first timestep of this tile
  const int bi  = blockIdx.y;               // batch index

  // Per-feature weights (K=4) and bias, one float4 of features per thread.
  const float4* W4 = (const float4*)W;
  const float4 w0 = W4[0 * F4 + tid];
  const float4 w1 = W4[1 * F4 + tid];
  const float4 w2 = W4[2 * F4 + tid];
  const float4 w3 = W4[3 * F4 + tid];
  const float4 bb = ((const float4*)bias)[tid];

  const unsigned rowbase = (unsigned)(bi * SEQ_LEN + t0);  // global row index
  const unsigned lane16  = (unsigned)tid * 16u;
  const unsigned long long xaddr = (unsigned long long)x;
  // addrspacecast(shared->flat) keeps the LDS byte offset in the low 32 bits.
  const unsigned ldsoff0 = (unsigned)(unsigned long long)(&lds[0]) + lane16;

  // Causal history x[t0-3..t0-1]: direct global loads (zeros at t==0).
  const float4* X4 = (const float4*)x;
  float4 h0, h1, h2;
  if (t0 == 0) {
    h0 = h1 = h2 = make_float4(0.f, 0.f, 0.f, 0.f);
  } else {
    const size_t hb = (size_t)rowbase * F4 + (size_t)tid;
    h0 = X4[hb - 3 * (size_t)F4];
    h1 = X4[hb - 2 * (size_t)F4];
    h2 = X4[hb - 1 * (size_t)F4];
  }

  // Per-lane VGPR byte offset of row t0; rows advance via imm offsets.
  unsigned goff = rowbase * (unsigned)ROW_BYTES + lane16;  // < 2^28, fits u32

  // Prologue: stage rows 0..7 into slots 0..7 (row offset in imm field).
#pragma unroll
  for (int j = 0; j < DEPTH; ++j)
    async_row_to_lds(xaddr, goff, ldsoff0 + (unsigned)(j * ROW_BYTES),
                     j * ROW_BYTES);

  float4* Y4g = (float4*)y + (size_t)rowbase * F4 + (size_t)tid;

  // Steady state: 7 groups of 8 rows. Unrolled x8 => slot index u and all
  // ds/store/async offsets are immediates. 8 async loads in flight; waiting
  // asynccnt<=7 guarantees (in-order retire) that row i has landed. After the
  // ds consume + s_wait_dscnt 0, the same slot is re-armed for row i+8.
  for (int g = 0; g < (T_TILE / DEPTH) - 1; ++g) {
#pragma unroll
    for (int u = 0; u < DEPTH; ++u) {
      WAIT_ASYNCCNT(DEPTH - 1);
      const float4 cur = lds[u * F4 + tid];        // ds_load_b128, imm offset

      float4 o = bb;
      o = fma4(w0, h0, o);
      o = fma4(w1, h1, o);
      o = fma4(w2, h2, o);
      o = fma4(w3, cur, o);
      store_nt(&Y4g[u * F4], o);                   // NT global_store_b128

      h0 = h1; h1 = h2; h2 = cur;

      WAIT_DSCNT(0);                               // slot data is in VGPRs
      async_row_to_lds(xaddr, goff, ldsoff0 + (unsigned)(u * ROW_BYTES),
                       (u + DEPTH) * ROW_BYTES);   // re-arm slot u with row+8
    }
    goff += (unsigned)(DEPTH * ROW_BYTES);
    Y4g  += DEPTH * F4;
  }

  // Drain: all 64 rows issued; the last 8 are resident after this wait.
  WAIT_ASYNCCNT(0);
#pragma unroll
  for (int u = 0; u < DEPTH; ++u) {
    const float4 cur = lds[u * F4 + tid];
    float4 o = bb;
    o = fma4(w0, h0, o);
    o = fma4(w1, h1, o);
    o = fma4(w2, h2, o);
    o = fma4(w3, cur, o);
    store_nt(&Y4g[u * F4], o);
    h0 = h1; h1 = h2; h2 = cur;
  }
}

extern "C" void kernel_launch(void* const* d_in, const int* in_sizes, int n_in,
                              void* d_out, int out_size, void* d_ws, size_t ws_size,
                              hipStream_t stream) {
  (void)in_sizes; (void)n_in; (void)out_size; (void)d_ws; (void)ws_size;
  const float* x = (const float*)d_in[0];
  const float* W = (const float*)d_in[1];
  const float* b = (const float*)d_in[2];
  float* y = (float*)d_out;

  dim3 grid(SEQ_LEN / T_TILE, BATCH);  // (128, 8) = 1024 workgroups
  dim3 block(F4);                      // 256 threads = 8 wave32
  hipLaunchKernelGGL(causal_dwconv4_kernel, grid, block, 0, stream, x, W, b, y);
}